// SimpleGNN_78649441124915
// MI455X (gfx1250) — compile-verified
//
#include <hip/hip_runtime.h>
#include <cstdint>

typedef __attribute__((ext_vector_type(2))) float v2f;
typedef __attribute__((ext_vector_type(8))) float v8f;

// ---------------------------------------------------------------------------
// Hardware fp32 atomic add (used only for the degree pass; E adds total).
// ---------------------------------------------------------------------------
__device__ __forceinline__ void atomAddF(float* p, float v) {
    unsafeAtomicAdd(p, v);
}

// ---------------------------------------------------------------------------
// Degree accumulation: deg[dst[e]] += ew[e]
// ---------------------------------------------------------------------------
__global__ void k_edge_deg(const long long* __restrict__ dst,
                           const float* __restrict__ ew,
                           float* __restrict__ deg, int E) {
    int e = blockIdx.x * blockDim.x + threadIdx.x;
    if (e < E) atomAddF(&deg[(int)dst[e]], ew[e]);
}

// deg -> dinv in place: dinv = rsqrt(deg + 1)  (self-loop weight 1)
__global__ void k_dinv(float* __restrict__ dd, int N) {
    int i = blockIdx.x * blockDim.x + threadIdx.x;
    if (i < N) {
        float d = dd[i] + 1.0f;
        dd[i] = (d > 0.0f) ? rsqrtf(d) : 0.0f;
    }
}

// ---------------------------------------------------------------------------
// CSR build by destination: counts -> exclusive scan -> fill.
// ---------------------------------------------------------------------------
__global__ void k_count(const long long* __restrict__ dst,
                        int* __restrict__ counts, int E) {
    int e = blockIdx.x * blockDim.x + threadIdx.x;
    if (e < E) atomicAdd(&counts[(int)dst[e]], 1);
}

// Per-256-block exclusive scan; block totals to bsum.
__global__ void k_scan_block(const int* __restrict__ in, int* __restrict__ out,
                             int* __restrict__ bsum, int N) {
    __shared__ int sm[256];
    int i = blockIdx.x * 256 + threadIdx.x;
    int v = (i < N) ? in[i] : 0;
    sm[threadIdx.x] = v;
    __syncthreads();
    for (int off = 1; off < 256; off <<= 1) {
        int t = (threadIdx.x >= off) ? sm[threadIdx.x - off] : 0;
        __syncthreads();
        sm[threadIdx.x] += t;
        __syncthreads();
    }
    int incl = sm[threadIdx.x];
    if (i < N) out[i] = incl - v;                 // exclusive
    if (threadIdx.x == 255) bsum[blockIdx.x] = incl;
}

// Single-block exclusive scan of up to 1024 block totals.
__global__ void k_scan_bsum(int* __restrict__ bsum, int nb) {
    __shared__ int sm[1024];
    int v = ((int)threadIdx.x < nb) ? bsum[threadIdx.x] : 0;
    sm[threadIdx.x] = v;
    __syncthreads();
    for (int off = 1; off < 1024; off <<= 1) {
        int t = ((int)threadIdx.x >= off) ? sm[threadIdx.x - off] : 0;
        __syncthreads();
        sm[threadIdx.x] += t;
        __syncthreads();
    }
    if ((int)threadIdx.x < nb) bsum[threadIdx.x] = sm[threadIdx.x] - v;
}

__global__ void k_scan_add(int* __restrict__ rowptr,
                           const int* __restrict__ bsum, int N, int E) {
    int i = blockIdx.x * blockDim.x + threadIdx.x;
    if (i < N) rowptr[i] += bsum[i >> 8];
    if (i == 0) rowptr[N] = E;
}

__global__ void k_copy_i32(const int* __restrict__ in, int* __restrict__ out,
                           int n) {
    int i = blockIdx.x * blockDim.x + threadIdx.x;
    if (i < n) out[i] = in[i];
}

// Scatter edges into CSR slots; also precompute norm = dinv[s]*ew*dinv[d].
__global__ void k_fill(const long long* __restrict__ src,
                       const long long* __restrict__ dst,
                       const float* __restrict__ ew,
                       const float* __restrict__ dinv,
                       int* __restrict__ fillptr, int* __restrict__ esrc_s,
                       float* __restrict__ enorm_s, int E) {
    int e = blockIdx.x * blockDim.x + threadIdx.x;
    if (e >= E) return;
    int s = (int)src[e];
    int d = (int)dst[e];
    int pos = atomicAdd(&fillptr[d], 1);
    esrc_s[pos] = s;
    enorm_s[pos] = dinv[s] * ew[e] * dinv[d];
}

// ---------------------------------------------------------------------------
// C[M,F] = A[M,K] * B[K,F] with V_WMMA_F32_16X16X4_F32 (full fp32 exactness;
// GEMM is far from the bottleneck so no reason to drop precision).
// One 32-lane wave per 16x16 tile. grid = (ceil(M/16), F/16), block = 32.
// Fragment layouts per CDNA5 ISA 7.12.2.
// ---------------------------------------------------------------------------
__global__ void k_gemm_wmma(const float* __restrict__ A,
                            const float* __restrict__ B,
                            float* __restrict__ C, int M, int K, int F) {
    const int lane = threadIdx.x & 31;
    const int half = lane >> 4;
    const int l16  = lane & 15;
    const int m0 = blockIdx.x * 16;
    const int n0 = blockIdx.y * 16;

    int arow = m0 + l16;
    if (arow >= M) arow = M - 1;     // clamp (keeps EXEC full; dup rows unused)

    const float* ap = A + (size_t)arow * K + 2 * half;
    const float* bp = B + (size_t)(2 * half) * F + n0 + l16;

    v8f acc = {};
    for (int k = 0; k < K; k += 4) {
        v2f a; a.x = ap[0]; a.y = ap[1];
        v2f b; b.x = bp[0]; b.y = bp[F];
        acc = __builtin_amdgcn_wmma_f32_16x16x4_f32(
            /*neg_a=*/false, a, /*neg_b=*/false, b,
            /*c_mod=*/(short)0, acc, /*reuse_a=*/false, /*reuse_b=*/false);
        ap += 4;
        bp += (size_t)4 * F;
    }
#pragma unroll
    for (int v = 0; v < 8; ++v) {
        int row = m0 + v + 8 * half;
        if (row < M) C[(size_t)row * F + n0 + l16] = acc[v];
    }
}

// ---------------------------------------------------------------------------
// CSR gather-aggregate, fused self-loop + bias + ReLU, zero atomics:
//   out[i] = relu( h[i]*dinv[i]^2 + sum_{j in row i} h[src[j]]*norm[j] + b )
// F/4 lanes cooperate per node (float4 per lane); h is L2-resident.
// ---------------------------------------------------------------------------
template <int F>
__global__ void k_aggregate(const int* __restrict__ rowptr,
                            const int* __restrict__ esrc,
                            const float* __restrict__ enorm,
                            const float* __restrict__ h,
                            const float* __restrict__ dinv,
                            const float* __restrict__ bias,
                            float* __restrict__ out, int N) {
    constexpr int G = F / 4;
    long long gid = (long long)blockIdx.x * blockDim.x + threadIdx.x;
    int i = (int)(gid / G);
    int c = (int)(gid % G) * 4;
    if (i >= N) return;

    float w = dinv[i];
    w *= w;
    float4 v = *(const float4*)(h + (size_t)i * F + c);
    float ax = v.x * w, ay = v.y * w, az = v.z * w, aw = v.w * w;

    int e0 = rowptr[i], e1 = rowptr[i + 1];
    for (int j = e0; j < e1; ++j) {
        int s = esrc[j];
        float nw = enorm[j];
        float4 hv = *(const float4*)(h + (size_t)s * F + c);
        ax = fmaf(hv.x, nw, ax);
        ay = fmaf(hv.y, nw, ay);
        az = fmaf(hv.z, nw, az);
        aw = fmaf(hv.w, nw, aw);
    }
    float4 o;
    o.x = fmaxf(ax + bias[c + 0], 0.0f);
    o.y = fmaxf(ay + bias[c + 1], 0.0f);
    o.z = fmaxf(az + bias[c + 2], 0.0f);
    o.w = fmaxf(aw + bias[c + 3], 0.0f);
    *(float4*)(out + (size_t)i * F + c) = o;
}

// ---------------------------------------------------------------------------
// Layer-3 aggregate (F=32) with the final FC fused:
//   t = relu(agg + b3);  out[i] = dot(t, Wfc) + bfc
// 8 lanes per node; cross-lane reduction via __shfl_xor within the group.
// ---------------------------------------------------------------------------
__global__ void k_agg_fc(const int* __restrict__ rowptr,
                         const int* __restrict__ esrc,
                         const float* __restrict__ enorm,
                         const float* __restrict__ h,
                         const float* __restrict__ dinv,
                         const float* __restrict__ b3,
                         const float* __restrict__ Wfc,
                         const float* __restrict__ bfc,
                         float* __restrict__ out, int N) {
    constexpr int F = 32, G = 8;
    long long gid = (long long)blockIdx.x * blockDim.x + threadIdx.x;
    int i = (int)(gid / G);
    int c = (int)(gid % G) * 4;
    if (i >= N) return;   // a whole 8-lane group maps to one node

    float w = dinv[i];
    w *= w;
    float4 v = *(const float4*)(h + (size_t)i * F + c);
    float ax = v.x * w, ay = v.y * w, az = v.z * w, aw = v.w * w;

    int e0 = rowptr[i], e1 = rowptr[i + 1];
    for (int j = e0; j < e1; ++j) {
        int s = esrc[j];
        float nw = enorm[j];
        float4 hv = *(const float4*)(h + (size_t)s * F + c);
        ax = fmaf(hv.x, nw, ax);
        ay = fmaf(hv.y, nw, ay);
        az = fmaf(hv.z, nw, az);
        aw = fmaf(hv.w, nw, aw);
    }
    ax = fmaxf(ax + b3[c + 0], 0.0f);
    ay = fmaxf(ay + b3[c + 1], 0.0f);
    az = fmaxf(az + b3[c + 2], 0.0f);
    aw = fmaxf(aw + b3[c + 3], 0.0f);

    float s4 = ax * Wfc[c + 0] + ay * Wfc[c + 1] + az * Wfc[c + 2] +
               aw * Wfc[c + 3];
    s4 += __shfl_xor(s4, 4, 32);
    s4 += __shfl_xor(s4, 2, 32);
    s4 += __shfl_xor(s4, 1, 32);
    if ((threadIdx.x & (G - 1)) == 0) out[i] = s4 + bfc[0];
}

// ---------------------------------------------------------------------------
extern "C" void kernel_launch(void* const* d_in, const int* in_sizes, int n_in,
                              void* d_out, int out_size, void* d_ws,
                              size_t ws_size, hipStream_t stream) {
    const float*     x    = (const float*)d_in[0];
    const long long* eidx = (const long long*)d_in[1];   // int64 per reference
    const float*     ew   = (const float*)d_in[2];
    const float*     W1   = (const float*)d_in[3];
    const float*     b1   = (const float*)d_in[4];
    const float*     W2   = (const float*)d_in[5];
    const float*     b2   = (const float*)d_in[6];
    const float*     W3   = (const float*)d_in[7];
    const float*     b3   = (const float*)d_in[8];
    const float*     Wfc  = (const float*)d_in[9];
    const float*     bfc  = (const float*)d_in[10];

    const int IN = 128, HID = 64, MID = 32;
    const int N = in_sizes[0] / IN;
    const int E = in_sizes[2];
    const long long* esrc64 = eidx;
    const long long* edst64 = eidx + E;

    // workspace layout (256B-aligned slices)
    char*  ws  = (char*)d_ws;
    size_t off = 0;
    auto carve = [&](size_t bytes) -> void* {
        void* p = ws + off;
        off += (bytes + 255) & ~(size_t)255;
        return p;
    };
    float* dinv    = (float*)carve((size_t)N * 4);
    int*   counts  = (int*)carve((size_t)N * 4);
    int*   rowptr  = (int*)carve((size_t)(N + 1) * 4);
    int*   fillptr = (int*)carve((size_t)N * 4);
    int*   bsum    = (int*)carve((size_t)1024 * 4);
    int*   esrc_s  = (int*)carve((size_t)E * 4);
    float* enorm_s = (float*)carve((size_t)E * 4);
    float* buf0    = (float*)carve((size_t)N * HID * 4);
    float* buf1    = (float*)carve((size_t)N * HID * 4);
    (void)ws_size; (void)n_in; (void)out_size;

    const int B = 256;
    auto cdiv = [](long long a, long long b) { return (int)((a + b - 1) / b); };
    const int nb = cdiv(N, 256);   // scan blocks (196 for N=50000, <=1024)

    // --- normalization + CSR build (once; reused by all 3 layers) ----------
    hipMemsetAsync(dinv, 0, (size_t)N * 4, stream);
    hipMemsetAsync(counts, 0, (size_t)N * 4, stream);
    k_edge_deg<<<cdiv(E, B), B, 0, stream>>>(edst64, ew, dinv, E);
    k_dinv<<<cdiv(N, B), B, 0, stream>>>(dinv, N);
    k_count<<<cdiv(E, B), B, 0, stream>>>(edst64, counts, E);
    k_scan_block<<<nb, 256, 0, stream>>>(counts, rowptr, bsum, N);
    k_scan_bsum<<<1, 1024, 0, stream>>>(bsum, nb);
    k_scan_add<<<cdiv(N, B), B, 0, stream>>>(rowptr, bsum, N, E);
    k_copy_i32<<<cdiv(N, B), B, 0, stream>>>(rowptr, fillptr, N);
    k_fill<<<cdiv(E, B), B, 0, stream>>>(esrc64, edst64, ew, dinv, fillptr,
                                         esrc_s, enorm_s, E);

    // --- layer 1: x[N,128] @ W1[128,64] -> aggregate+bias+relu -------------
    k_gemm_wmma<<<dim3(cdiv(N, 16), HID / 16), 32, 0, stream>>>(x, W1, buf0, N, IN, HID);
    k_aggregate<64><<<cdiv((long long)N * 16, B), B, 0, stream>>>(
        rowptr, esrc_s, enorm_s, buf0, dinv, b1, buf1, N);

    // --- layer 2: h1[N,64] @ W2[64,64] -> aggregate+bias+relu --------------
    k_gemm_wmma<<<dim3(cdiv(N, 16), HID / 16), 32, 0, stream>>>(buf1, W2, buf0, N, HID, HID);
    k_aggregate<64><<<cdiv((long long)N * 16, B), B, 0, stream>>>(
        rowptr, esrc_s, enorm_s, buf0, dinv, b2, buf1, N);

    // --- layer 3: h2[N,64] @ W3[64,32] -> aggregate+bias+relu+FC -----------
    k_gemm_wmma<<<dim3(cdiv(N, 16), MID / 16), 32, 0, stream>>>(buf1, W3, buf0, N, HID, MID);
    k_agg_fc<<<cdiv((long long)N * 8, B), B, 0, stream>>>(
        rowptr, esrc_s, enorm_s, buf0, dinv, b3, Wfc, bfc, (float*)d_out, N);
}